// DarkTrafficAttentionDetectorLoss_5738076308213
// MI455X (gfx1250) — compile-verified
//
#include <hip/hip_runtime.h>
#include <math.h>

#define BB 32
#define PP 42840
#define OO 8
#define NNI 8
#define CC 11
#define THRESH 0.4f
#define THETA_C 0.1f

typedef __attribute__((ext_vector_type(2))) float v2f;
typedef __attribute__((ext_vector_type(8))) float v8f;

// ---------------------------------------------------------------------------
// Async global->LDS copy (ASYNCcnt-tracked) per CDNA5 ISA.
// VDST = VGPR holding LDS byte address, VADDR = 64-bit global address pair.
// ---------------------------------------------------------------------------
__device__ __forceinline__ void async_load16(unsigned lds_off, const void* gaddr) {
  asm volatile("global_load_async_to_lds_b128 %0, %1, off"
               :: "v"(lds_off), "v"(gaddr) : "memory");
}
__device__ __forceinline__ void wait_async() {
  asm volatile("s_wait_asynccnt 0x0" ::: "memory");
}

// ---------------------------------------------------------------------------
// Wave-level sum via V_WMMA_F32_16X16X4_F32 with an all-ones B matrix.
// A per-lane = {v, 0}:  A(m,0)=v_m (lanes 0-15), A(m,2)=v_{m+16} (lanes 16-31)
// => D[m][n] = v_m + v_{m+16} for every n.  Sum the 8 accumulator regs,
// then one xor-16 shuffle.  Requires all 32 lanes active -> call uniformly.
// ---------------------------------------------------------------------------
__device__ __forceinline__ float wave_reduce_wmma(float v) {
  v2f a; a.x = v; a.y = 0.0f;
  v2f ones; ones.x = 1.0f; ones.y = 1.0f;
  v8f c = {};
  c = __builtin_amdgcn_wmma_f32_16x16x4_f32(false, a, false, ones,
                                            (short)0, c, false, false);
  float s = c[0] + c[1] + c[2] + c[3] + c[4] + c[5] + c[6] + c[7];
  s += __shfl_xor(s, 16, 32);
  return s;  // full 32-lane sum in every lane
}

// Block sum (blockDim.x <= 256, multiple of 32). Result valid on thread 0.
__device__ float block_reduce(float v) {
  __shared__ float lds8[8];
  float w = wave_reduce_wmma(v);
  int lane = threadIdx.x & 31, wid = threadIdx.x >> 5;
  if (lane == 0) lds8[wid] = w;
  __syncthreads();
  float r = 0.f;
  if (threadIdx.x == 0) {
    int nw = (blockDim.x + 31) >> 5;
    for (int i = 0; i < nw; ++i) r += lds8[i];
  }
  __syncthreads();
  return r;
}

__device__ __forceinline__ float iou_corner(float ax0, float ay0, float ax1, float ay1,
                                            float bx0, float by0, float bx1, float by1) {
  float lx = fmaxf(ax0, bx0), ly = fmaxf(ay0, by0);
  float rx = fminf(ax1, bx1), ry = fminf(ay1, by1);
  float w = fmaxf(rx - lx, 0.f), h = fmaxf(ry - ly, 0.f);
  float inter = w * h;
  float aa = (ax1 - ax0) * (ay1 - ay0);
  float ab = (bx1 - bx0) * (by1 - by0);
  return inter / (aa + ab - inter);
}

// ---------------------------------------------------------------------------
__global__ void k_init(float* acc, int* npos) {
  int t = threadIdx.x;
  if (t < 16) acc[t] = 0.f;
  if (t < BB) npos[t] = 0;
}

// seg loss: sum(-max(log1p(-x), -100)) over B*56*96 = 172032 (exact grid)
__global__ void k_seg(const float* __restrict__ att, float* acc) {
  int i = blockIdx.x * blockDim.x + threadIdx.x;
  float x = att[i];
  float l = fmaxf(log1pf(-x), -100.f);
  float s = block_reduce(-l);
  if (threadIdx.x == 0) atomicAdd(&acc[0], s);
}

// per (b,o): max IoU over priors + first-occurrence argmax
__global__ void k_objmatch(const float* __restrict__ boxes,
                           const float* __restrict__ priors,
                           float* __restrict__ ovobj, int* __restrict__ pobj) {
  int b = blockIdx.x;
  __shared__ float sb[OO * 4];
  if (threadIdx.x < OO * 4) sb[threadIdx.x] = boxes[b * OO * 4 + threadIdx.x];
  __syncthreads();
  float bestv[OO]; int besti[OO];
#pragma unroll
  for (int o = 0; o < OO; ++o) { bestv[o] = -1.f; besti[o] = 0; }
  for (int p = threadIdx.x; p < PP; p += blockDim.x) {
    float cx = priors[p * 4 + 0], cy = priors[p * 4 + 1];
    float w  = priors[p * 4 + 2], h  = priors[p * 4 + 3];
    float x0 = cx - w * 0.5f, y0 = cy - h * 0.5f;
    float x1 = cx + w * 0.5f, y1 = cy + h * 0.5f;
#pragma unroll
    for (int o = 0; o < OO; ++o) {
      float v = iou_corner(sb[o * 4], sb[o * 4 + 1], sb[o * 4 + 2], sb[o * 4 + 3],
                           x0, y0, x1, y1);
      if (v > bestv[o]) { bestv[o] = v; besti[o] = p; }  // strict > keeps first
    }
  }
  __shared__ float rv[256];
  __shared__ int   ri[256];
  for (int o = 0; o < OO; ++o) {
    rv[threadIdx.x] = bestv[o]; ri[threadIdx.x] = besti[o];
    __syncthreads();
    for (int s = blockDim.x >> 1; s > 0; s >>= 1) {
      if ((int)threadIdx.x < s) {
        float v2 = rv[threadIdx.x + s]; int i2 = ri[threadIdx.x + s];
        if (v2 > rv[threadIdx.x] ||
            (v2 == rv[threadIdx.x] && i2 < ri[threadIdx.x])) {
          rv[threadIdx.x] = v2; ri[threadIdx.x] = i2;
        }
      }
      __syncthreads();
    }
    if (threadIdx.x == 0) { ovobj[b * OO + o] = rv[0]; pobj[b * OO + o] = ri[0]; }
    __syncthreads();
  }
}

// fused per-prior: matching + decode + DIoU + CE; writes conf_neg bits
__global__ void k_main(const float* __restrict__ locs, const float* __restrict__ scores,
                       const float* __restrict__ boxes, const int* __restrict__ labels,
                       const float* __restrict__ ign, const float* __restrict__ priors,
                       const float* __restrict__ ovobj, const int* __restrict__ pobj,
                       unsigned* __restrict__ confb, float* acc, int* npos) {
  int b = blockIdx.y;
  int p = blockIdx.x * blockDim.x + threadIdx.x;
  // [0..31] = boxes row, [32..63] = ignored-regions row; filled by async DMA
  __shared__ __align__(16) float sconst[64];
  __shared__ int   sl[OO];
  __shared__ float sov[OO];
  __shared__ int   spo[OO];
  if (threadIdx.x < 16) {
    const float* src = (threadIdx.x < 8)
        ? &boxes[(size_t)b * OO * 4 + threadIdx.x * 4]
        : &ign[(size_t)b * NNI * 4 + (threadIdx.x - 8) * 4];
    async_load16((unsigned)(size_t)&sconst[threadIdx.x * 4], src);
  }
  if (threadIdx.x < OO) {
    sl[threadIdx.x]  = labels[b * OO + threadIdx.x];
    sov[threadIdx.x] = ovobj[b * OO + threadIdx.x];
    spo[threadIdx.x] = pobj[b * OO + threadIdx.x];
  }
  wait_async();       // drain this wave's ASYNCcnt before the barrier
  __syncthreads();
  const float* sb = &sconst[0];
  const float* sg = &sconst[32];
  bool act = p < PP;
  float loc_c = 0.f, confpos_c = 0.f, pos_c = 0.f;
  if (act) {
    __builtin_prefetch(&scores[((size_t)b * PP + p) * CC], 0, 1);  // global_prefetch_b8
    float cx = priors[p * 4 + 0], cy = priors[p * 4 + 1];
    float pw = priors[p * 4 + 2], ph = priors[p * 4 + 3];
    float x0 = cx - pw * 0.5f, y0 = cy - ph * 0.5f;
    float x1 = cx + pw * 0.5f, y1 = cy + ph * 0.5f;
    float ovp = -1.f; int am = 0;
#pragma unroll
    for (int o = 0; o < OO; ++o) {
      float v = iou_corner(sb[o * 4], sb[o * 4 + 1], sb[o * 4 + 2], sb[o * 4 + 3],
                           x0, y0, x1, y1);
      if (v > ovp) { ovp = v; am = o; }
    }
    float ig = -1.f;
#pragma unroll
    for (int i = 0; i < NNI; ++i)
      ig = fmaxf(ig, iou_corner(sg[i * 4], sg[i * 4 + 1], sg[i * 4 + 2], sg[i * 4 + 3],
                                x0, y0, x1, y1));
    // filtered-rank "last write wins" override (reference semantics)
    int assigned = -1; bool over = false; int r = -1;
#pragma unroll
    for (int o = 0; o < OO; ++o) {
      bool valid = sov[o] > 0.f;
      if (valid) {
        ++r;
        if (spo[o] == p) { over = true; if (r > assigned) assigned = r; }
      }
    }
    int obj = (assigned >= 0) ? assigned : am;
    float ovfin = over ? 1.f : ovp;
    int label = sl[obj];
    if (ovfin < THRESH) label = 0;
    bool pos  = label > 0;
    bool ignf = ig >= THETA_C;
    // decode gcxgcy -> cxcy -> corners
    const float* g = &locs[((size_t)b * PP + p) * 4];
    float dcx = g[0] * pw * 0.1f + cx;
    float dcy = g[1] * ph * 0.1f + cy;
    float dw  = expf(g[2] * 0.2f) * pw;
    float dh  = expf(g[3] * 0.2f) * ph;
    float px0 = dcx - dw * 0.5f, py0 = dcy - dh * 0.5f;
    float px1 = dcx + dw * 0.5f, py1 = dcy + dh * 0.5f;
    float gx0 = sb[obj * 4], gy0 = sb[obj * 4 + 1];
    float gx1 = sb[obj * 4 + 2], gy1 = sb[obj * 4 + 3];
    // DIoU loss
    float lx = fmaxf(px0, gx0), ly = fmaxf(py0, gy0);
    float rx = fminf(px1, gx1), ry = fminf(py1, gy1);
    float iw = fmaxf(rx - lx, 0.f), ih = fmaxf(ry - ly, 0.f);
    float inter = iw * ih;
    float ap = (px1 - px0) * (py1 - py0), ag = (gx1 - gx0) * (gy1 - gy0);
    float iou = inter / (ap + ag - inter);
    float cpx = (px0 + px1) * 0.5f, cpy = (py0 + py1) * 0.5f;
    float cgx = (gx0 + gx1) * 0.5f, cgy = (gy0 + gy1) * 0.5f;
    float idg = (cpx - cgx) * (cpx - cgx) + (cpy - cgy) * (cpy - cgy);
    float ox0 = fminf(px0, gx0), oy0 = fminf(py0, gy0);
    float ox1 = fmaxf(px1, gx1), oy1 = fmaxf(py1, gy1);
    float odg = (ox1 - ox0) * (ox1 - ox0) + (oy1 - oy0) * (oy1 - oy0);
    float diou = fminf(fmaxf(iou - idg / odg, -1.f), 1.f);
    float lloss = 1.f - diou;
    // stable log-softmax CE
    const float* sc = &scores[((size_t)b * PP + p) * CC];
    float m = sc[0];
#pragma unroll
    for (int k = 1; k < CC; ++k) m = fmaxf(m, sc[k]);
    float se = 0.f;
#pragma unroll
    for (int k = 0; k < CC; ++k) se += expf(sc[k] - m);
    float conf = (m + logf(se)) - sc[label];
    if (pos) { loc_c = lloss; confpos_c = conf; pos_c = 1.f; }
    float cn = (pos || ignf) ? 0.f : fmaxf(conf, 0.f);  // >=0 -> monotone bits
    confb[(size_t)b * PP + p] = __float_as_uint(cn);
  }
  float s;
  s = block_reduce(loc_c);
  if (threadIdx.x == 0 && s != 0.f) atomicAdd(&acc[1], s);
  s = block_reduce(confpos_c);
  if (threadIdx.x == 0 && s != 0.f) atomicAdd(&acc[2], s);
  s = block_reduce(pos_c);
  if (threadIdx.x == 0) {
    int n = (int)(s + 0.5f);
    if (n) atomicAdd(&npos[b], n);
  }
}

// exact top-K sum per batch row via 4-round byte radix-select (no sort)
__global__ void k_topk(const unsigned* __restrict__ confb,
                       const int* __restrict__ npos, float* acc) {
  int b = blockIdx.x;
  const unsigned* row = confb + (size_t)b * PP;
  int K = 2 * npos[b];
  if (K > PP) K = PP;
  if (K <= 0) return;  // uniform per block
  __shared__ int hist[256];
  __shared__ int s_sel, s_rem;
  unsigned prefix = 0u, pmask = 0u;
  int remaining = K;
  for (int shift = 24; shift >= 0; shift -= 8) {
    for (int i = threadIdx.x; i < 256; i += blockDim.x) hist[i] = 0;
    __syncthreads();
    for (int p = threadIdx.x; p < PP; p += blockDim.x) {
      unsigned v = row[p];
      if ((v & pmask) == prefix) atomicAdd(&hist[(v >> shift) & 0xFFu], 1);
    }
    __syncthreads();
    if (threadIdx.x == 0) {
      int cum = 0, bin = 255;
      for (; bin >= 0; --bin) { cum += hist[bin]; if (cum >= remaining) break; }
      if (bin < 0) bin = 0;
      s_sel = bin;
      s_rem = remaining - (cum - hist[bin]);
    }
    __syncthreads();
    prefix |= ((unsigned)s_sel) << shift;
    pmask  |= 0xFFu << shift;
    remaining = s_rem;
    __syncthreads();
  }
  // sum of strictly-greater + ties at the K-th value
  float sum = 0.f;
  for (int p = threadIdx.x; p < PP; p += blockDim.x) {
    unsigned v = row[p];
    if (v > prefix) sum += __uint_as_float(v);
  }
  float bs = block_reduce(sum);
  if (threadIdx.x == 0)
    atomicAdd(&acc[3], bs + (float)remaining * __uint_as_float(prefix));
}

__global__ void k_final(const float* acc, const int* npos, float* out) {
  if (threadIdx.x == 0) {
    float tp = 0.f;
    for (int b = 0; b < BB; ++b) tp += (float)npos[b];
    float conf = (acc[3] + acc[2]) / tp;
    float loc  = acc[1] / tp;
    out[0] = conf + loc + acc[0];
  }
}

// ---------------------------------------------------------------------------
extern "C" void kernel_launch(void* const* d_in, const int* in_sizes, int n_in,
                              void* d_out, int out_size, void* d_ws, size_t ws_size,
                              hipStream_t stream) {
  const float* odm_locs   = (const float*)d_in[0];
  const float* odm_scores = (const float*)d_in[1];
  const float* attention  = (const float*)d_in[2];
  const float* boxes      = (const float*)d_in[3];
  const int*   labels     = (const int*)d_in[4];
  const float* ignored    = (const float*)d_in[5];
  const float* priors     = (const float*)d_in[6];
  float* out = (float*)d_out;

  float* wsf      = (float*)d_ws;
  float* acc      = wsf;                       // 16 floats
  int*   npos     = (int*)(wsf + 16);          // 32 ints
  float* ovobj    = wsf + 64;                  // B*O floats
  int*   pobj     = (int*)(wsf + 320);         // B*O ints
  unsigned* confb = (unsigned*)(wsf + 1024);   // B*P uints (~5.5 MB)

  k_init<<<1, 64, 0, stream>>>(acc, npos);
  k_seg<<<672, 256, 0, stream>>>(attention, acc);            // 672*256 == 172032
  k_objmatch<<<BB, 256, 0, stream>>>(boxes, priors, ovobj, pobj);
  dim3 g3((PP + 255) / 256, BB);
  k_main<<<g3, 256, 0, stream>>>(odm_locs, odm_scores, boxes, labels, ignored,
                                 priors, ovobj, pobj, confb, acc, npos);
  k_topk<<<BB, 256, 0, stream>>>(confb, npos, acc);
  k_final<<<1, 32, 0, stream>>>(acc, npos, out);
}